// Attention_66889820668421
// MI455X (gfx1250) — compile-verified
//
#include <hip/hip_runtime.h>

typedef __attribute__((ext_vector_type(2))) float v2f;
typedef __attribute__((ext_vector_type(8))) float v8f;

#define B_ 8
#define H_ 16
#define M_ 1024
#define N_ 1024
#define D_ 64
#define TILE_M 16
#define WAVES 4
// 1024 + 4 floats of pad: row stride of 1028 DWORDs => row r and row r+8
// differ by 8*1028 = 32 (mod 64) banks -> no half-wave bank conflicts.
#define LDS_STRIDE 1028

__global__ __launch_bounds__(WAVES * 32) void attn_fp32_wmma(
    const float* __restrict__ q, const float* __restrict__ k,
    const float* __restrict__ v, const int* __restrict__ mask,
    float* __restrict__ out, float* __restrict__ p_attn)
{
  extern __shared__ float smem[];          // TILE_M * LDS_STRIDE floats (shared by 4 waves)

  const int lane = threadIdx.x & 31;       // wave32 lane
  const int wave = threadIdx.x >> 5;       // 0..3
  const int half = lane >> 4;              // 0: K even pair / rows 0-7; 1: K odd pair / rows 8-15
  const int l15  = lane & 15;

  const int    mt = blockIdx.x;            // M tile: 0..63
  const size_t bh = blockIdx.y;            // fused (b,h): 0..127
  const int    m0 = mt * TILE_M;

  const float scale = 0.03125f;            // 1/sqrt(N) = 1/sqrt(1024), faithful to reference

  const float* qbh = q + (bh * M_ + m0) * (size_t)D_;
  const float* kbh = k + bh * (size_t)N_ * D_;
  const float* vbh = v + bh * (size_t)N_ * D_;
  const int*   mbh = mask + bh * N_;

  // ---- Q tile (16x64) -> 16 fp32 A-fragments, kept in VGPRs (per wave) ----
  // A 16x4 f32 layout: lanes 0-15 hold (M=lane, K=0,1); lanes 16-31 (M=lane-16, K=2,3)
  v2f qa[16];
  const float* qrow = qbh + (size_t)l15 * D_ + 2 * half;
  #pragma unroll
  for (int kk = 0; kk < 16; ++kk)
    qa[kk] = *(const v2f*)(qrow + 4 * kk);

  // ---- phase 1: scores = (Q K^T)*scale with mask, into LDS ----------------
  // Wave w covers N-tiles [16w, 16w+16); two tiles per iteration -> two
  // independent WMMA accumulation chains (ILP) on top of 4-wave occupancy.
  for (int j = wave * 16; j < wave * 16 + 16; j += 2) {
    const int n0 = j * TILE_M;
    // B 4x16 f32 layout: lanes 0-15 hold (K=0,1 ; N=lane); lanes 16-31 (K=2,3 ; N=lane-16)
    const float* krow0 = kbh + (size_t)(n0 + l15) * D_ + 2 * half;
    const float* krow1 = krow0 + (size_t)TILE_M * D_;
    v8f c0 = {}; v8f c1 = {};
    #pragma unroll
    for (int kk = 0; kk < 16; ++kk) {
      v2f b0 = *(const v2f*)(krow0 + 4 * kk);
      v2f b1 = *(const v2f*)(krow1 + 4 * kk);
      c0 = __builtin_amdgcn_wmma_f32_16x16x4_f32(false, qa[kk], false, b0,
                                                 (short)0, c0, false, false);
      c1 = __builtin_amdgcn_wmma_f32_16x16x4_f32(false, qa[kk], false, b1,
                                                 (short)0, c1, false, false);
    }
    // D 16x16 f32 layout: VGPR r, lanes 0-15 -> (M=r, N=lane); lanes 16-31 -> (M=r+8, N=lane-16)
    const int  n_a = n0 + l15;
    const int  n_b = n_a + TILE_M;
    const bool kp0 = mbh[n_a] != 0;
    const bool kp1 = mbh[n_b] != 0;
    #pragma unroll
    for (int r = 0; r < 8; ++r) {
      const int m = r + 8 * half;
      smem[m * LDS_STRIDE + n_a] = kp0 ? c0[r] * scale : -1e9f;
      smem[m * LDS_STRIDE + n_b] = kp1 ? c1[r] * scale : -1e9f;
    }
  }
  __syncthreads();

  // ---- phase 2: row softmax; write normalized P to LDS and p_attn ---------
  // Wave w handles rows [4w, 4w+4); 32 lanes cooperate per row.
  float* pbase = p_attn + (bh * M_ + m0) * (size_t)N_;
  for (int m = wave * 4; m < wave * 4 + 4; ++m) {
    float* srow = &smem[m * LDS_STRIDE];
    float mx = -3.4e38f;
    for (int c = lane; c < N_; c += 32) mx = fmaxf(mx, srow[c]);
    #pragma unroll
    for (int off = 16; off >= 1; off >>= 1) mx = fmaxf(mx, __shfl_xor(mx, off, 32));
    float sum = 0.0f;
    for (int c = lane; c < N_; c += 32) {
      float e = __expf(srow[c] - mx);
      srow[c] = e;
      sum += e;
    }
    #pragma unroll
    for (int off = 16; off >= 1; off >>= 1) sum += __shfl_xor(sum, off, 32);
    const float inv = 1.0f / sum;
    float* prow = pbase + (size_t)m * N_;
    for (int c = lane; c < N_; c += 32) {   // coalesced 128B stores
      float pv = srow[c] * inv;
      srow[c] = pv;
      prow[c] = pv;
    }
  }
  __syncthreads();

  // ---- phase 3: out = P (16x1024) x V (1024x64) ---------------------------
  // Wave w owns D-tile dt = w over the full K range; 4 interleaved
  // accumulator chains (kk mod 4) keep WMMA ILP, summed in-register at end.
  const int dbase = wave * 16 + l15;
  v8f acc[4] = {{}, {}, {}, {}};
  for (int kk4 = 0; kk4 < N_ / 16; ++kk4) {
    #pragma unroll
    for (int i = 0; i < 4; ++i) {
      const int k0 = (kk4 * 4 + i) * 4;
      // A-fragment of P from LDS (pad stride keeps this conflict-free)
      v2f a = *(const v2f*)(&smem[l15 * LDS_STRIDE + k0 + 2 * half]);
      // B 4x16 f32 layout from V rows k0..k0+3, columns dbase
      const float* vr = vbh + (size_t)(k0 + 2 * half) * D_ + dbase;
      v2f bb;
      bb.x = vr[0];
      bb.y = vr[D_];
      acc[i] = __builtin_amdgcn_wmma_f32_16x16x4_f32(false, a, false, bb,
                                                     (short)0, acc[i], false, false);
    }
  }
  v8f accs = (acc[0] + acc[1]) + (acc[2] + acc[3]);

  float* obase = out + (bh * M_ + m0) * (size_t)D_;
  #pragma unroll
  for (int r = 0; r < 8; ++r) {
    const int m = r + 8 * half;
    obase[(size_t)m * D_ + dbase] = accs[r];
  }
}

extern "C" void kernel_launch(void* const* d_in, const int* in_sizes, int n_in,
                              void* d_out, int out_size, void* d_ws, size_t ws_size,
                              hipStream_t stream) {
  (void)in_sizes; (void)n_in; (void)out_size; (void)d_ws; (void)ws_size;
  const float* q    = (const float*)d_in[0];
  const float* k    = (const float*)d_in[1];
  const float* v    = (const float*)d_in[2];
  const int*   mask = (const int*)d_in[3];

  float* out    = (float*)d_out;                       // [B,H,M,D]
  float* p_attn = out + (size_t)B_ * H_ * M_ * D_;     // [B,H,M,N], concatenated

  dim3 grid(M_ / TILE_M, B_ * H_);
  dim3 block(WAVES * 32);
  size_t lds_bytes = (size_t)TILE_M * LDS_STRIDE * sizeof(float);  // ~64.3 KB / WG
  hipLaunchKernelGGL(attn_fp32_wmma, grid, block, lds_bytes, stream,
                     q, k, v, mask, out, p_attn);
}